// GraphSAGEEncoder_20701742366801
// MI455X (gfx1250) — compile-verified
//
#include <hip/hip_runtime.h>
#include <math.h>

#define IN_DIM 7
#define HID 128

typedef __attribute__((ext_vector_type(2))) float v2f;
typedef __attribute__((ext_vector_type(8))) float v8f;

static __device__ __forceinline__ void atomicAddF32(float* p, float v) {
    __hip_atomic_fetch_add(p, v, __ATOMIC_RELAXED, __HIP_MEMORY_SCOPE_AGENT);
}

// ---------------------------------------------------------------------------
// Kernel 1: per-edge degree count + 7-dim feature scatter-add (layer 0 agg)
// ---------------------------------------------------------------------------
__global__ void sage_edge_agg0(const long long* __restrict__ ei,
                               const float* __restrict__ x,
                               float* __restrict__ cnt,
                               float* __restrict__ agg0,   // [N,8] padded
                               long long E) {
    long long e = (long long)blockIdx.x * blockDim.x + threadIdx.x;
    if (e >= E) return;
    long long src = ei[e];
    long long dst = ei[E + e];
    atomicAddF32(&cnt[dst], 1.0f);
    const float* xs = x + src * IN_DIM;
    float* a = agg0 + dst * 8;
#pragma unroll
    for (int k = 0; k < IN_DIM; ++k) atomicAddF32(&a[k], xs[k]);
}

// ---------------------------------------------------------------------------
// Kernel 2: layer-0 dense: h = relu(mean0 @ Wl0^T + bl0 + x @ Wr0^T)
// one block per node, thread j = output channel (K=7 -> scalar FMA is right)
// ---------------------------------------------------------------------------
__global__ void sage_layer0(const float* __restrict__ x,
                            const float* __restrict__ agg0,
                            const float* __restrict__ cnt,
                            const float* __restrict__ Wl0,
                            const float* __restrict__ bl0,
                            const float* __restrict__ Wr0,
                            float* __restrict__ h, int n) {
    int i = blockIdx.x;
    if (i >= n) return;
    int j = threadIdx.x;
    __shared__ float sx[8];
    __shared__ float sm[8];
    __shared__ float sinv;
    if (j < IN_DIM) sx[j] = x[(size_t)i * IN_DIM + j];
    if (j >= 8 && j < 8 + IN_DIM) sm[j - 8] = agg0[(size_t)i * 8 + (j - 8)];
    if (j == 15) sinv = 1.0f / fmaxf(cnt[i], 1.0f);
    __syncthreads();
    float inv = sinv;
    float acc = bl0[j];
    const float* wl = Wl0 + j * IN_DIM;
    const float* wr = Wr0 + j * IN_DIM;
#pragma unroll
    for (int k = 0; k < IN_DIM; ++k)
        acc += (sm[k] * inv) * wl[k] + sx[k] * wr[k];
    h[(size_t)i * HID + j] = fmaxf(acc, 0.0f);
}

// ---------------------------------------------------------------------------
// Kernel 3: layer-1 aggregation. One wave32 per edge, each lane handles 4 dims.
// h and agg1 are both L2-resident (51 MB each, 192 MB L2).
// ---------------------------------------------------------------------------
__global__ void sage_edge_agg1(const long long* __restrict__ ei,
                               const float* __restrict__ h,
                               float* __restrict__ agg1,
                               long long E) {
    int lane = threadIdx.x & 31;
    long long e = (long long)blockIdx.x * 8 + (threadIdx.x >> 5);
    if (e >= E) return;
    long long src = ei[e];
    long long dst = ei[E + e];
    const float4 hv = *(const float4*)(h + src * HID + lane * 4);
    float* a = agg1 + dst * HID + lane * 4;
    atomicAddF32(&a[0], hv.x);
    atomicAddF32(&a[1], hv.y);
    atomicAddF32(&a[2], hv.z);
    atomicAddF32(&a[3], hv.w);
}

// ---------------------------------------------------------------------------
// Kernel 4: layer-1 dense via V_WMMA_F32_16X16X4_F32.
// out[16 nodes][128] = [mean1|h] (16x256, LDS) x [Wl1^T;Wr1^T] (256x128) + bl1
// Block = 256 thr = 8 waves; wave w owns columns [16w,16w+16). In-place on h.
// A layout (16x4 f32): lanes 0-15 row M=lane K={k0,k0+1}; lanes 16-31 K={k0+2,k0+3}.
// B layout (4x16): lane serves column ncol; b = {B[k0+kofs][ncol], B[k0+kofs+1][ncol]}
//                  with B[k][n] = W[n][k]  (row-major W -> contiguous float2).
// C/D layout: lane -> col = wave*16 + (lane&15); VGPR r -> row r + 8*(lane>>4).
// ---------------------------------------------------------------------------
__global__ void sage_layer1_wmma(const float* __restrict__ agg1,
                                 const float* __restrict__ cnt,
                                 const float* __restrict__ Wl1,
                                 const float* __restrict__ bl1,
                                 const float* __restrict__ Wr1,
                                 float* __restrict__ h,   // in: layer-0 act, out: result
                                 int n) {
    const int LDA = 2 * HID + 2;            // 258: even (b64-aligned), not mult of 64
    __shared__ float As[16 * LDA];
    __shared__ float sInv[16];

    const int tid  = threadIdx.x;
    const int base = blockIdx.x * 16;

    if (tid < 16) {
        int node = base + tid;
        sInv[tid] = (node < n) ? (1.0f / fmaxf(cnt[node], 1.0f)) : 0.0f;
    }
    __syncthreads();

    // Stage A = [mean1 | h] for 16 rows (all global reads of our rows happen here)
    for (int idx = tid; idx < 16 * 2 * HID; idx += 256) {
        int row  = idx >> 8;                 // 0..15
        int col  = idx & 255;                // 0..255
        int node = base + row;
        float v = 0.0f;
        if (node < n) {
            if (col < HID) v = agg1[(size_t)node * HID + col] * sInv[row];
            else           v = h[(size_t)node * HID + (col - HID)];
        }
        As[row * LDA + col] = v;
    }
    __syncthreads();

    const int wave = tid >> 5;
    const int lane = tid & 31;
    const int half = lane >> 4;             // 0: K+0..1, 1: K+2..3
    const int arow = lane & 15;
    const int ncol = wave * 16 + arow;      // output column this lane services
    const int kofs = half * 2;

    const float* wlp = Wl1 + (size_t)ncol * HID + kofs;
    const float* wrp = Wr1 + (size_t)ncol * HID + kofs;

    v8f acc = {0.f, 0.f, 0.f, 0.f, 0.f, 0.f, 0.f, 0.f};

#pragma unroll 8
    for (int k0 = 0; k0 < HID; k0 += 4) {   // mean1 @ Wl1^T
        v2f a = *(const v2f*)&As[arow * LDA + kofs + k0];
        v2f b = *(const v2f*)(wlp + k0);
        acc = __builtin_amdgcn_wmma_f32_16x16x4_f32(false, a, false, b,
                                                    (short)0, acc, false, false);
    }
#pragma unroll 8
    for (int k0 = 0; k0 < HID; k0 += 4) {   // h @ Wr1^T
        v2f a = *(const v2f*)&As[arow * LDA + HID + kofs + k0];
        v2f b = *(const v2f*)(wrp + k0);
        acc = __builtin_amdgcn_wmma_f32_16x16x4_f32(false, a, false, b,
                                                    (short)0, acc, false, false);
    }

    float bias = bl1[ncol];
#pragma unroll
    for (int r = 0; r < 8; ++r) {
        int node = base + r + half * 8;
        if (node < n) h[(size_t)node * HID + ncol] = acc[r] + bias;
    }
}

// ---------------------------------------------------------------------------
extern "C" void kernel_launch(void* const* d_in, const int* in_sizes, int n_in,
                              void* d_out, int out_size, void* d_ws, size_t ws_size,
                              hipStream_t stream) {
    const float*     x   = (const float*)d_in[0];
    const long long* ei  = (const long long*)d_in[1];   // int64 [2,E]
    const float*     Wl0 = (const float*)d_in[2];
    const float*     bl0 = (const float*)d_in[3];
    const float*     Wr0 = (const float*)d_in[4];
    const float*     Wl1 = (const float*)d_in[5];
    const float*     bl1 = (const float*)d_in[6];
    const float*     Wr1 = (const float*)d_in[7];

    const int       N = in_sizes[0] / IN_DIM;
    const long long E = in_sizes[1] / 2;

    // Workspace: cnt[N] | agg0[N*8] | agg1[N*128]   (h lives in d_out)
    float* ws   = (float*)d_ws;
    float* cnt  = ws;
    float* agg0 = ws + (size_t)N;
    float* agg1 = ws + (size_t)N * 9;
    float* h    = (float*)d_out;

    hipMemsetAsync(d_ws, 0, sizeof(float) * (size_t)N * 137, stream);

    {   // layer-0 aggregation
        dim3 grid((unsigned)((E + 255) / 256));
        sage_edge_agg0<<<grid, 256, 0, stream>>>(ei, x, cnt, agg0, E);
    }
    {   // layer-0 dense + relu -> h (in d_out)
        sage_layer0<<<N, HID, 0, stream>>>(x, agg0, cnt, Wl0, bl0, Wr0, h, N);
    }
    {   // layer-1 aggregation (wave per edge)
        dim3 grid((unsigned)((E + 7) / 8));
        sage_edge_agg1<<<grid, 256, 0, stream>>>(ei, h, agg1, E);
    }
    {   // layer-1 dense via fp32 WMMA, in-place on d_out
        dim3 grid((unsigned)((N + 15) / 16));
        sage_layer1_wmma<<<grid, 256, 0, stream>>>(agg1, cnt, Wl1, bl1, Wr1, h, N);
    }
}